// IGMC_68229850464741
// MI455X (gfx1250) — compile-verified
//
#include <hip/hip_runtime.h>
#include <hip/hip_bf16.h>

// ---------------------------------------------------------------------------
// IGMC (4-layer basis-decomposed RGCN + MLP head) for MI455X / gfx1250.
//
// Strategy:
//  * hb[n,b,:] = h @ basis_b   -> fp32 WMMA (V_WMMA_F32_16X16X4_F32), 50 MB,
//    lives in L2 (192 MB) during the edge scatter.
//  * per-edge: m = c0*hb0[src] + c1*hb1[src]; atomic-add into S2[dst,rel,:]
//    (126 MB, also L2-resident).  Counts per (dst,rel) computed once.
//  * combine: h_new = tanh(h@root + bias + sum_r S2[n,r,:]*inv_cnt[n,r]),
//    draining+zeroing S2 for the next layer.
//  * head: feat=[user||item] (2048x256) @ w1 -> relu -> @ w2, via WMMA.
// ---------------------------------------------------------------------------

typedef float v2f __attribute__((ext_vector_type(2)));
typedef float v8f __attribute__((ext_vector_type(8)));

#define RNUM 5

// ---------------- zero scratch -------------------------------------------
__global__ void zero_f(float* __restrict__ p, size_t n) {
    size_t i = (size_t)blockIdx.x * blockDim.x + threadIdx.x;
    size_t stride = (size_t)gridDim.x * blockDim.x;
    for (; i < n; i += stride) p[i] = 0.0f;
}

// ---------------- per-(dst,rel) edge counts ------------------------------
__global__ __launch_bounds__(256) void count_edges(const int* __restrict__ dst,
                                                   const int* __restrict__ et,
                                                   float* __restrict__ cnt, int E) {
    int e = blockIdx.x * blockDim.x + threadIdx.x;
    if (e >= E) return;
    atomicAdd(cnt + (size_t)dst[e] * RNUM + et[e], 1.0f);
}

__global__ __launch_bounds__(256) void invert_cnt(float* __restrict__ cnt, int n) {
    int i = blockIdx.x * blockDim.x + threadIdx.x;
    if (i >= n) return;
    cnt[i] = 1.0f / fmaxf(cnt[i], 1.0f);
}

// ---------------- dense: D[16 x 32] = A[16 x K] @ W[K x 32] (+bias) -------
// One wave per 16-row tile, two 16x16 fp32 WMMA accumulators (cols 0-15/16-31).
// ISA layouts (CDNA5 ISA §7.12.2):
//   A lane l, vgpr j : A[l%16, k0 + 2*(l/16) + j]
//   B lane l, vgpr j : B[k0 + 2*(l/16) + j, l%16]
//   D lane l, vgpr j : D[j + 8*(l/16), l%16]
__global__ __launch_bounds__(256) void dense_nk32(const float* __restrict__ A, int lda,
                                                  const float* __restrict__ W, int K,
                                                  float* __restrict__ D, int ldd,
                                                  const float* __restrict__ bias, int nrows) {
    int wave = threadIdx.x >> 5;
    int lane = threadIdx.x & 31;
    int row0 = (blockIdx.x * 8 + wave) * 16;
    if (row0 >= nrows) return;
    int m = lane & 15, hi = lane >> 4;

    const float* arow = A + (size_t)(row0 + m) * lda + 2 * hi;
    v8f acc0 = {0.f,0.f,0.f,0.f,0.f,0.f,0.f,0.f};
    v8f acc1 = {0.f,0.f,0.f,0.f,0.f,0.f,0.f,0.f};

    for (int k0 = 0; k0 < K; k0 += 4) {
        int kb = k0 + 2 * hi;
        v2f a;  a.x = arow[k0];           a.y = arow[k0 + 1];
        v2f b0; b0.x = W[kb * 32 + m];        b0.y = W[(kb + 1) * 32 + m];
        v2f b1; b1.x = W[kb * 32 + 16 + m];   b1.y = W[(kb + 1) * 32 + 16 + m];
        acc0 = __builtin_amdgcn_wmma_f32_16x16x4_f32(false, a, false, b0, (short)0, acc0, false, false);
        acc1 = __builtin_amdgcn_wmma_f32_16x16x4_f32(false, a, false, b1, (short)0, acc1, false, false);
    }

    float bi0 = bias ? bias[m] : 0.0f;
    float bi1 = bias ? bias[16 + m] : 0.0f;
#pragma unroll
    for (int j = 0; j < 8; ++j) {
        int r = row0 + j + 8 * hi;
        D[(size_t)r * ldd + m]       = acc0[j] + bi0;
        D[(size_t)r * ldd + 16 + m]  = acc1[j] + bi1;
    }
}

// ---------------- edge scatter: S2[dst,rel,:] += c0*hb0[src]+c1*hb1[src] --
__global__ __launch_bounds__(256) void scatter_edges(const int* __restrict__ src,
                                                     const int* __restrict__ dst,
                                                     const int* __restrict__ et,
                                                     const float* __restrict__ comp,  // [R,2]
                                                     const float* __restrict__ hb,    // [N,64]
                                                     float* __restrict__ S2, int E) {
    int tid = blockIdx.x * blockDim.x + threadIdx.x;
    int e = tid >> 3;
    if (e >= E) return;
    int j0 = (tid & 7) * 4;
    int s = src[e], d = dst[e], r = et[e];
    float c0 = comp[r * 2], c1 = comp[r * 2 + 1];
    const float4 v0 = *(const float4*)(hb + (size_t)s * 64 + j0);
    const float4 v1 = *(const float4*)(hb + (size_t)s * 64 + 32 + j0);
    float* o = S2 + ((size_t)d * RNUM + r) * 32 + j0;
    atomicAdd(o + 0, c0 * v0.x + c1 * v1.x);
    atomicAdd(o + 1, c0 * v0.y + c1 * v1.y);
    atomicAdd(o + 2, c0 * v0.z + c1 * v1.z);
    atomicAdd(o + 3, c0 * v0.w + c1 * v1.w);
}

// ---------------- combine: tanh(root_pre + sum_r S2*inv) ; drain S2 -------
__global__ __launch_bounds__(256) void combine_layer(float* __restrict__ S2,
                                                     const float* __restrict__ inv,
                                                     float* __restrict__ states,
                                                     int colOff, int N) {
    int tid = blockIdx.x * blockDim.x + threadIdx.x;
    if (tid >= N * 32) return;
    int n = tid >> 5, j = tid & 31;
    float* s2 = S2 + (size_t)n * (RNUM * 32) + j;
    const float* iv = inv + (size_t)n * RNUM;
    float a = 0.0f;
#pragma unroll
    for (int r = 0; r < RNUM; ++r) {
        a += s2[r * 32] * iv[r];
        s2[r * 32] = 0.0f;          // re-zero for next layer's scatter
    }
    size_t idx = (size_t)n * 128 + colOff + j;
    states[idx] = tanhf(states[idx] + a);
}

// ---------------- MLP1: hidden = relu([user||item] @ w1 + b1) via WMMA ----
__global__ __launch_bounds__(256) void mlp1_wmma(const float* __restrict__ states,
                                                 const int* __restrict__ uidx,
                                                 const int* __restrict__ iidx,
                                                 const float* __restrict__ w1,  // [256,128]
                                                 const float* __restrict__ b1,  // [128]
                                                 float* __restrict__ hidden, int G) {
    int wave = threadIdx.x >> 5;
    int lane = threadIdx.x & 31;
    int wid = blockIdx.x * 8 + wave;
    int row0 = (wid >> 2) * 16;       // 16-row tile
    int cb = wid & 3;                 // 32-col block
    if (row0 >= G) return;
    int m = lane & 15, hi = lane >> 4;
    int row = row0 + m;
    const float* ru = states + (size_t)uidx[row] * 128;
    const float* ri = states + (size_t)iidx[row] * 128;
    int c0 = cb * 32 + m;

    v8f acc0 = {0.f,0.f,0.f,0.f,0.f,0.f,0.f,0.f};
    v8f acc1 = {0.f,0.f,0.f,0.f,0.f,0.f,0.f,0.f};

    for (int k0 = 0; k0 < 256; k0 += 4) {
        int k = k0 + 2 * hi;
        const float* srow = (k < 128) ? (ru + k) : (ri + (k - 128));
        v2f a;  a.x = srow[0];  a.y = srow[1];
        v2f b0; b0.x = w1[(size_t)k * 128 + c0];       b0.y = w1[(size_t)(k + 1) * 128 + c0];
        v2f b1v; b1v.x = w1[(size_t)k * 128 + c0 + 16]; b1v.y = w1[(size_t)(k + 1) * 128 + c0 + 16];
        acc0 = __builtin_amdgcn_wmma_f32_16x16x4_f32(false, a, false, b0,  (short)0, acc0, false, false);
        acc1 = __builtin_amdgcn_wmma_f32_16x16x4_f32(false, a, false, b1v, (short)0, acc1, false, false);
    }

    float bi0 = b1[c0], bi1 = b1[c0 + 16];
#pragma unroll
    for (int j = 0; j < 8; ++j) {
        int r = row0 + j + 8 * hi;
        hidden[(size_t)r * 128 + c0]      = fmaxf(acc0[j] + bi0, 0.0f);
        hidden[(size_t)r * 128 + c0 + 16] = fmaxf(acc1[j] + bi1, 0.0f);
    }
}

// ---------------- MLP2: out = hidden @ w2 + b2 (one wave per row) ---------
__global__ __launch_bounds__(256) void mlp2_dot(const float* __restrict__ hidden,
                                                const float* __restrict__ w2,  // [128]
                                                const float* __restrict__ b2,
                                                float* __restrict__ out, int G) {
    int wave = threadIdx.x >> 5;
    int lane = threadIdx.x & 31;
    int g = blockIdx.x * 8 + wave;
    if (g >= G) return;
    const float4 h4 = *(const float4*)(hidden + (size_t)g * 128 + lane * 4);
    const float4 w4 = *(const float4*)(w2 + lane * 4);
    float s = h4.x * w4.x + h4.y * w4.y + h4.z * w4.z + h4.w * w4.w;
#pragma unroll
    for (int off = 16; off > 0; off >>= 1) s += __shfl_down(s, off, 32);
    if (lane == 0) out[g] = s + b2[0];
}

// ---------------------------------------------------------------------------
extern "C" void kernel_launch(void* const* d_in, const int* in_sizes, int n_in,
                              void* d_out, int out_size, void* d_ws, size_t ws_size,
                              hipStream_t stream) {
    const float* x      = (const float*)d_in[0];
    const int*   eidx   = (const int*)d_in[1];
    const int*   etype  = (const int*)d_in[2];
    const int*   uidx   = (const int*)d_in[3];
    const int*   iidx   = (const int*)d_in[4];
    const float* basis0 = (const float*)d_in[5];   // [2,4,32]
    const float* comp0  = (const float*)d_in[6];   // [R,2]
    const float* root0  = (const float*)d_in[7];   // [4,32]
    const float* bias0  = (const float*)d_in[8];   // [32]
    const float* basisR = (const float*)d_in[9];   // [3,2,32,32]
    const float* compR  = (const float*)d_in[10];  // [3,R,2]
    const float* rootR  = (const float*)d_in[11];  // [3,32,32]
    const float* biasR  = (const float*)d_in[12];  // [3,32]
    const float* w1     = (const float*)d_in[13];  // [256,128]
    const float* b1     = (const float*)d_in[14];  // [128]
    const float* w2     = (const float*)d_in[15];  // [128,1]
    const float* b2     = (const float*)d_in[16];  // [1]
    float* out = (float*)d_out;

    const int N = in_sizes[0] / 4;
    const int E = in_sizes[2];
    const int G = in_sizes[3];
    const int* srcp = eidx;
    const int* dstp = eidx + E;

    // workspace layout (floats): states[N,128] | hb[N,64] | S2[N,R,32] | cnt[N,R] | hidden[G,128]
    float* ws     = (float*)d_ws;
    float* states = ws;
    float* hb     = states + (size_t)N * 128;
    float* S2     = hb + (size_t)N * 64;
    float* cnt    = S2 + (size_t)N * (RNUM * 32);
    float* hidden = cnt + (size_t)N * RNUM;

    // zero S2 + cnt (contiguous), every call (harness does not re-poison)
    zero_f<<<4096, 256, 0, stream>>>(S2, (size_t)N * (RNUM * 32 + RNUM));

    // per-(dst,rel) counts -> reciprocals (edge structure identical for all layers)
    count_edges<<<(E + 255) / 256, 256, 0, stream>>>(dstp, etype, cnt, E);
    invert_cnt<<<(N * RNUM + 255) / 256, 256, 0, stream>>>(cnt, N * RNUM);

    const int denseBlocks = ((N + 15) / 16 + 7) / 8;
    const int scatBlocks  = (E * 8 + 255) / 256;
    const int combBlocks  = (N * 32 + 255) / 256;

    // ---- layer 0 (input x: [N,4], K=4) ----
    dense_nk32<<<denseBlocks, 256, 0, stream>>>(x, 4, basis0,        4, hb,      64, nullptr, N);
    dense_nk32<<<denseBlocks, 256, 0, stream>>>(x, 4, basis0 + 128,  4, hb + 32, 64, nullptr, N);
    dense_nk32<<<denseBlocks, 256, 0, stream>>>(x, 4, root0,         4, states, 128, bias0,   N);
    scatter_edges<<<scatBlocks, 256, 0, stream>>>(srcp, dstp, etype, comp0, hb, S2, E);
    combine_layer<<<combBlocks, 256, 0, stream>>>(S2, cnt, states, 0, N);

    // ---- layers 1..3 (h: [N,32] slice of states, K=32) ----
    for (int i = 0; i < 3; ++i) {
        const float* hin = states + 32 * i;
        dense_nk32<<<denseBlocks, 256, 0, stream>>>(hin, 128, basisR + (size_t)(2 * i) * 1024,     32, hb,      64, nullptr, N);
        dense_nk32<<<denseBlocks, 256, 0, stream>>>(hin, 128, basisR + (size_t)(2 * i + 1) * 1024, 32, hb + 32, 64, nullptr, N);
        dense_nk32<<<denseBlocks, 256, 0, stream>>>(hin, 128, rootR + (size_t)i * 1024,            32,
                                                    states + 32 * (i + 1), 128, biasR + i * 32, N);
        scatter_edges<<<scatBlocks, 256, 0, stream>>>(srcp, dstp, etype, compR + i * RNUM * 2, hb, S2, E);
        combine_layer<<<combBlocks, 256, 0, stream>>>(S2, cnt, states, 32 * (i + 1), N);
    }

    // ---- MLP head ----
    int mlp1Blocks = (((G + 15) / 16) * 4 + 7) / 8;
    mlp1_wmma<<<mlp1Blocks, 256, 0, stream>>>(states, uidx, iidx, w1, b1, hidden, G);
    mlp2_dot<<<(G + 7) / 8, 256, 0, stream>>>(hidden, w2, b2, out, G);
}